// SOM_29824252903830
// MI455X (gfx1250) — compile-verified
//
#include <hip/hip_runtime.h>

// ---------------------------------------------------------------------------
// SOM update for MI455X (gfx1250, wave32).
//
// Roofline: weights = 256 MB. Read once (distance pass), read+write (update
// pass) => ~768 MB => ~33 us at 23.3 TB/s HBM. Compute is ~0.5 GFLOP, i.e.
// pure bandwidth problem. Kernels are structured as streams; argmin uses one
// u64 atomic-min per block (512 total), exp() once per row.
//
// CDNA5 path: per-row sum(diff^2) via V_WMMA_F32_16X16X4_F32 as diag(A*A^T).
// For the f32 16x4 A layout and 4x16 B layout in the ISA tables, A and A^T
// have the SAME per-lane register image (lane<16: row=lane, K={0,1};
// lane>=16: row=lane-16, K={2,3}), so one contiguous b64 load per lane feeds
// both WMMA operands. (Layout assumption documented; no runtime check here.)
// ---------------------------------------------------------------------------

typedef __attribute__((ext_vector_type(2))) float v2f;
typedef __attribute__((ext_vector_type(8))) float v8f;

#define SOM_DIM   1024
#define SOM_EPS   1e-6f
#define ROWS_PER_WAVE 16
#define ROWS_PER_BLOCK 128   // 8 waves * 16 rows

__global__ void som_init_key(unsigned long long* key) {
    *key = 0xFFFFFFFFFFFFFFFFull;
}

// One wave handles 16 rows. K-loop: 4 dims per WMMA, unrolled x2 with two
// accumulators to break the D->C dependency chain.
__global__ __launch_bounds__(256)
void som_dist_argmin(const float* __restrict__ x,
                     const float* __restrict__ w,
                     unsigned long long* __restrict__ key) {
    __shared__ float cspill[8 * 256];        // per-wave 16x16 C matrix spill
    __shared__ unsigned long long blockMin;

    const int lane = threadIdx.x & 31;
    const int wave = threadIdx.x >> 5;
    const int rowBase = blockIdx.x * ROWS_PER_BLOCK + wave * ROWS_PER_WAVE;
    const int m  = lane & 15;     // row within 16-row tile
    const int hi = lane >> 4;     // 0 -> K pair {0,1}, 1 -> K pair {2,3}

    if (threadIdx.x == 0) blockMin = 0xFFFFFFFFFFFFFFFFull;

    const float* wrow = w + (size_t)(rowBase + m) * SOM_DIM + hi * 2;
    const float* xp   = x + hi * 2;

    v8f c0 = {};
    v8f c1 = {};
    for (int k = 0; k < SOM_DIM; k += 8) {
        v2f w0 = *(const v2f*)(wrow + k);
        v2f x0 = *(const v2f*)(xp   + k);
        v2f w1 = *(const v2f*)(wrow + k + 4);
        v2f x1 = *(const v2f*)(xp   + k + 4);
        v2f a0, a1;
        a0.x = x0.x - w0.x + SOM_EPS;
        a0.y = x0.y - w0.y + SOM_EPS;
        a1.x = x1.x - w1.x + SOM_EPS;
        a1.y = x1.y - w1.y + SOM_EPS;
        // diag(C) accumulates sum of squared diffs per row; A == A^T in regs.
        c0 = __builtin_amdgcn_wmma_f32_16x16x4_f32(
                 false, a0, false, a0, (short)0, c0, false, false);
        c1 = __builtin_amdgcn_wmma_f32_16x16x4_f32(
                 false, a1, false, a1, (short)0, c1, false, false);
    }
    v8f c = c0 + c1;

    // Spill C to LDS, pull the diagonal: D[m][m] lives at
    //   m<8 : vgpr m,   lane m
    //   m>=8: vgpr m-8, lane m+16
    float* cl = &cspill[wave * 256];
    #pragma unroll
    for (int r = 0; r < 8; ++r) cl[r * 32 + lane] = c[r];
    __syncthreads();

    unsigned long long my = 0xFFFFFFFFFFFFFFFFull;
    if (lane < 16) {
        float d2 = (lane < 8) ? cl[lane * 32 + lane]
                              : cl[(lane - 8) * 32 + (lane + 16)];
        // d2 >= 0 so its IEEE bit pattern is order-preserving as u32.
        // Low 32 bits = row index => ties resolve to lowest index (argmin).
        unsigned int bits = __float_as_uint(d2);
        my = ((unsigned long long)bits << 32) |
             (unsigned long long)(unsigned int)(rowBase + lane);
    }
    #pragma unroll
    for (int off = 16; off > 0; off >>= 1) {
        unsigned long long o = __shfl_down(my, off, 32);
        my = (o < my) ? o : my;
    }
    if (lane == 0) atomicMin(&blockMin, my);   // ds_min_u64
    __syncthreads();
    if (threadIdx.x == 0) atomicMin(key, blockMin);  // global_atomic_min_u64
}

// One block per row: 256 threads * float4 = 1024 elems. Pure stream.
__global__ __launch_bounds__(256)
void som_update(const float* __restrict__ x,
                const float* __restrict__ w,
                const int* __restrict__ loc,
                const int* __restrict__ itp,
                const unsigned long long* __restrict__ key,
                float* __restrict__ out,
                int n_rows) {
    __shared__ float sfactor;
    const int row = blockIdx.x;

    if (threadIdx.x == 0) {
        const int bmu = (int)(unsigned int)(*key & 0xFFFFFFFFull);
        const float itf = (float)(*itp);
        const float lr = 1.0f - itf * 0.01f;          // 1 - it/NITER
        const float alpha_op = 0.3f * lr;
        const float sigma_op = 32768.0f * lr;
        const float d = (float)(loc[row] - bmu);
        sfactor = alpha_op * __expf(-(d * d) / (sigma_op * sigma_op));
    }

    const size_t base = (size_t)row * SOM_DIM;
    const float4* wr = (const float4*)(w + base);
    const float4* xr = (const float4*)x;
    float4* o = (float4*)(out + base);
    const int t = threadIdx.x;

    // Speculative streaming prefetch a few rows ahead (global_prefetch_b8).
    if (row + 8 < n_rows)
        __builtin_prefetch(w + base + (size_t)8 * SOM_DIM + t * 4, 0, 0);

    float4 wv = wr[t];
    float4 xv = xr[t];
    __syncthreads();
    const float f = sfactor;

    float4 r;
    r.x = wv.x + f * (xv.x - wv.x);
    r.y = wv.y + f * (xv.y - wv.y);
    r.z = wv.z + f * (xv.z - wv.z);
    r.w = wv.w + f * (xv.w - wv.w);
    o[t] = r;
}

extern "C" void kernel_launch(void* const* d_in, const int* in_sizes, int n_in,
                              void* d_out, int out_size, void* d_ws, size_t ws_size,
                              hipStream_t stream) {
    const float* x   = (const float*)d_in[0];     // (1024,)
    const float* w   = (const float*)d_in[1];     // (65536, 1024)
    const int*   loc = (const int*)d_in[2];       // (65536,)
    const int*   itp = (const int*)d_in[3];       // scalar it
    float* out = (float*)d_out;
    unsigned long long* key = (unsigned long long*)d_ws;

    const int N = in_sizes[2];                    // 65536 rows

    som_init_key<<<1, 1, 0, stream>>>(key);
    som_dist_argmin<<<N / ROWS_PER_BLOCK, 256, 0, stream>>>(x, w, key);
    som_update<<<N, 256, 0, stream>>>(x, w, loc, itp, key, out, N);
}